// SimpleGCNNet_46316927320539
// MI455X (gfx1250) — compile-verified
//
#include <hip/hip_runtime.h>

typedef __attribute__((ext_vector_type(2))) float v2f;
typedef __attribute__((ext_vector_type(4))) float v4f;
typedef __attribute__((ext_vector_type(8))) float v8f;

constexpr int T_DIM = 256;   // input feature dim
constexpr int H_DIM = 128;   // output feature dim

static inline size_t alignup(size_t v, size_t a) { return (v + a - 1) & ~(a - 1); }

// deg[i] = 1.0 (self-loop weight)
__global__ void k_init_deg(float* __restrict__ deg, int N) {
  int i = blockIdx.x * blockDim.x + threadIdx.x;
  if (i < N) deg[i] = 1.0f;
}

// deg[col[e]] += clamp(w[e])
__global__ void k_deg_scatter(const long long* __restrict__ ei,
                              const float* __restrict__ w,
                              float* __restrict__ deg, int E) {
  int e = blockIdx.x * blockDim.x + threadIdx.x;
  if (e < E) {
    float ww = w[e];
    ww = (ww <= 0.0f) ? 1e-7f : ww;            // elu(w)<=0 <=> w<=0
    int c = (int)ei[(long long)E + e];         // edge_index[1]
    unsafeAtomicAdd(&deg[c], ww);
  }
}

// dis = deg>0 ? rsqrt(deg) : 0  (in place)
__global__ void k_rsqrt(float* __restrict__ deg, int N) {
  int i = blockIdx.x * blockDim.x + threadIdx.x;
  if (i < N) {
    float d = deg[i];
    deg[i] = (d > 0.0f) ? rsqrtf(d) : 0.0f;
  }
}

// norm[e] = dis[row]*clamp(w)*dis[col]
__global__ void k_norm(const long long* __restrict__ ei,
                       const float* __restrict__ w,
                       const float* __restrict__ dis,
                       float* __restrict__ norm, int E) {
  int e = blockIdx.x * blockDim.x + threadIdx.x;
  if (e < E) {
    float ww = w[e];
    ww = (ww <= 0.0f) ? 1e-7f : ww;
    int r = (int)ei[e];
    int c = (int)ei[(long long)E + e];
    norm[e] = dis[r] * ww * dis[c];
  }
}

// Y[N,128] = X[N,256] @ W[128,256]^T via V_WMMA_F32_16X16X4_F32.
// One wave per 16-row stripe; 8 N-tiles (full 128 cols); 64 K-steps of 4.
// Requires N % 16 == 0 (true: 50000 = 3125*16) so EXEC is all ones.
__global__ __launch_bounds__(32) void k_gemm_wmma(const float* __restrict__ X,
                                                  const float* __restrict__ Wm,
                                                  float* __restrict__ Y) {
  const int lane  = threadIdx.x;           // 0..31
  const int l15   = lane & 15;
  const int khalf = (lane >> 4) << 1;      // lanes 0-15 -> K 0,1 ; lanes 16-31 -> K 2,3
  const size_t mrow = (size_t)blockIdx.x * 16 + l15;

  v8f acc[8] = {};
  const float* xrow = X + mrow * T_DIM + khalf;

  for (int k = 0; k < T_DIM; k += 4) {
    v2f a = *(const v2f*)(xrow + k);       // A fragment: 16x4 f32
#pragma unroll
    for (int nt = 0; nt < 8; ++nt) {
      const int ncol = nt * 16 + l15;
      v2f bfrag = *(const v2f*)(Wm + (size_t)ncol * T_DIM + k + khalf); // B = W^T
      acc[nt] = __builtin_amdgcn_wmma_f32_16x16x4_f32(
          /*neg_a=*/false, a, /*neg_b=*/false, bfrag,
          /*c_mod=*/(short)0, acc[nt], /*reuse_a=*/false, /*reuse_b=*/false);
    }
  }

  // D layout: VGPR v -> row v (lanes 0-15) or v+8 (lanes 16-31), col = lane&15
  const size_t rbase = (size_t)blockIdx.x * 16 + ((lane >> 4) << 3);
#pragma unroll
  for (int nt = 0; nt < 8; ++nt) {
    const int ncol = nt * 16 + l15;
#pragma unroll
    for (int v = 0; v < 8; ++v) {
      Y[(rbase + v) * H_DIM + ncol] = acc[nt][v];
    }
  }
}

// yout[i,t..t+3] = dis[i]^2 * yin[i,t..t+3] (+ bias) -- self-loop term, float4
__global__ void k_self_scale4(const float* __restrict__ dis,
                              const float* __restrict__ yin,
                              float* __restrict__ yout,
                              const float* __restrict__ bias, int N) {
  int idx = blockIdx.x * blockDim.x + threadIdx.x;   // N*32 threads
  if (idx < N * (H_DIM / 4)) {
    int i = idx >> 5;            // node  (32 float4 per row)
    int q = (idx & 31) << 2;     // feature start
    float d = dis[i];
    float s = d * d;
    v4f y = *(const v4f*)(yin + (size_t)i * H_DIM + q);
    v4f v;
    v.x = s * y.x; v.y = s * y.y; v.z = s * y.z; v.w = s * y.w;
    if (bias) {
      v4f bb = *(const v4f*)(bias + q);
      v.x += bb.x; v.y += bb.y; v.z += bb.z; v.w += bb.w;
    }
    *(v4f*)(yout + (size_t)i * H_DIM + q) = v;
  }
}

// scatter-add over edges: yout[col,:] += norm[e] * yin[row,:]
// one thread per (edge, 4 features): float4 gather (global_load_b128) + 4
// hardware f32 atomics. A wave (32 lanes x 4 feats = 128) covers exactly one
// edge, so row/col/norm loads are wave-uniform and atomics hit one cacheline
// range.
__global__ void k_prop_edges4(const long long* __restrict__ ei,
                              const float* __restrict__ norm,
                              const float* __restrict__ yin,
                              float* __restrict__ yout, int E) {
  int idx = blockIdx.x * blockDim.x + threadIdx.x;   // E*32 = 25.6M
  if (idx < E * (H_DIM / 4)) {
    int e = idx >> 5;            // edge (wave-uniform)
    int q = (idx & 31) << 2;     // feature start
    int r = (int)ei[e];
    int c = (int)ei[(long long)E + e];
    float nw = norm[e];
    v4f y = *(const v4f*)(yin + (size_t)r * H_DIM + q);
    float* dst = yout + (size_t)c * H_DIM + q;
    unsafeAtomicAdd(dst + 0, nw * y.x);
    unsafeAtomicAdd(dst + 1, nw * y.y);
    unsafeAtomicAdd(dst + 2, nw * y.z);
    unsafeAtomicAdd(dst + 3, nw * y.w);
  }
}

extern "C" void kernel_launch(void* const* d_in, const int* in_sizes, int n_in,
                              void* d_out, int out_size, void* d_ws, size_t ws_size,
                              hipStream_t stream) {
  const float*     x  = (const float*)d_in[0];      // [N, 256]
  const long long* ei = (const long long*)d_in[1];  // [2, E] int64
  const float*     ew = (const float*)d_in[2];      // [E]
  const float*     Wm = (const float*)d_in[3];      // [128, 256]
  const float*     b  = (const float*)d_in[4];      // [128]
  float* out = (float*)d_out;                       // [N, 128]

  const int N = in_sizes[0] / T_DIM;                // 50000
  const int E = in_sizes[2];                        // 800000

  // workspace: dis[N] | norm[E] | y0[N*128] | y1[N*128]  (~55 MB)
  char* ws = (char*)d_ws;
  size_t off = 0;
  float* dis  = (float*)(ws + off); off += alignup((size_t)N * 4, 256);
  float* norm = (float*)(ws + off); off += alignup((size_t)E * 4, 256);
  float* y0   = (float*)(ws + off); off += alignup((size_t)N * H_DIM * 4, 256);
  float* y1   = (float*)(ws + off);

  const dim3 blk(256);

  // gcn_norm
  k_init_deg   <<<dim3((N + 255) / 256), blk, 0, stream>>>(dis, N);
  k_deg_scatter<<<dim3((E + 255) / 256), blk, 0, stream>>>(ei, ew, dis, E);
  k_rsqrt      <<<dim3((N + 255) / 256), blk, 0, stream>>>(dis, N);
  k_norm       <<<dim3((E + 255) / 256), blk, 0, stream>>>(ei, ew, dis, norm, E);

  // y0 = X @ W^T  (GEMM first: P^2(XW^T) == (P^2 X)W^T, halves propagation cost)
  k_gemm_wmma<<<dim3(N / 16), dim3(32), 0, stream>>>(x, Wm, y0);

  const int nv = N * (H_DIM / 4);   // float4 elements per dense pass
  const int ev = E * (H_DIM / 4);   // (edge, float4) work items

  // hop 1: y1 = P y0
  k_self_scale4<<<dim3((nv + 255) / 256), blk, 0, stream>>>(dis, y0, y1, nullptr, N);
  k_prop_edges4<<<dim3((ev + 255) / 256), blk, 0, stream>>>(ei, norm, y0, y1, E);

  // hop 2: out = P y1 + b
  k_self_scale4<<<dim3((nv + 255) / 256), blk, 0, stream>>>(dis, y1, out, b, N);
  k_prop_edges4<<<dim3((ev + 255) / 256), blk, 0, stream>>>(ei, norm, y1, out, E);
}